// ConvSubsetRouting_76699525972648
// MI455X (gfx1250) — compile-verified
//
#include <hip/hip_runtime.h>
#include <stdint.h>

// Problem dims (fixed by the reference)
#define B_ 8
#define I_ 32
#define O_ 32
#define D_ 8
#define H_ 14
#define W_ 14
#define K_ 9
#define WK_ (W_*K_)                  // 126 contiguous floats per (b,i,o,d,h)
#define CHI_ 8                       // input caps per TDM chunk
#define NCH_ (I_/CHI_)               // 4 chunks
#define SLAB_ (CHI_*D_*WK_)          // 8064 floats per chunk (32 KB)
#define STRIDE_D_ (H_*W_*K_)         // 1764   (element stride of d)
#define STRIDE_I_ (O_*D_*H_*W_*K_)   // 451584 (element stride of i)
#define SUBSET_ 26                   // ceil(0.8 * 32)

typedef uint32_t u32;
typedef uint32_t v4u __attribute__((ext_vector_type(4)));
typedef int      v8i __attribute__((ext_vector_type(8)));
typedef int      v4i __attribute__((ext_vector_type(4)));

// Issue one TDM 3D-tile load: 126 contiguous floats (dim0) x 8 d-lines
// (stride 1764 elems, dim1) x 8 i-lines (stride 451584 elems, dim2),
// written densely into LDS at lds_byte. D# layout per CDNA5 ISA ch.8.
__device__ __forceinline__ void tdm_load_tile(const float* gsrc, u32 lds_byte) {
  uint64_t ga = (uint64_t)(uintptr_t)gsrc;            // byte address
  v4u g0;
  g0[0] = 1u;                                          // count=1, user mode, no gather
  g0[1] = lds_byte;                                    // LDS dest (bytes, wg-relative)
  g0[2] = (u32)ga;                                     // global_addr[31:0]
  g0[3] = (u32)((ga >> 32) & 0x1FFFFFFull) | (2u << 30); // global_addr[56:32], type=2
  v8i g1;
  g1[0] = (int)(2u << 16);                             // data_size=2 (4 bytes)
  g1[1] = (int)((u32)WK_ << 16);                       // tensor_dim0 = 126
  g1[2] = (int)((u32)D_  << 16);                       // tensor_dim1 = 8
  g1[3] = (int)((u32)WK_ << 16);                       // tile_dim0   = 126
  g1[4] = (int)(((u32)CHI_ << 16) | (u32)D_);          // tile_dim1=8, tile_dim2=8
  g1[5] = (int)STRIDE_D_;                              // tensor_dim0_stride lo32 = 1764
  g1[6] = (int)(((u32)STRIDE_I_ & 0xFFFFu) << 16);     // dim0_stride hi16=0 | dim1_stride lo16
  g1[7] = (int)((u32)STRIDE_I_ >> 16);                 // dim1_stride [47:16]
  v4i g2; g2[0] = CHI_; g2[1] = 0; g2[2] = 0; g2[3] = 0; // tensor_dim2 = 8
  v4i g3; g3[0] = 0;    g3[1] = 0; g3[2] = 0; g3[3] = 0;
  v8i g4; g4[0]=0; g4[1]=0; g4[2]=0; g4[3]=0; g4[4]=0; g4[5]=0; g4[6]=0; g4[7]=0;
  __builtin_amdgcn_tensor_load_to_lds(g0, g1, g2, g3, g4, 0);
}

__global__ __launch_bounds__(256)
void ConvSubsetRouting_76699525972648_kernel(const float* __restrict__ x,
                                             float* __restrict__ out) {
  __shared__ float slab[2][SLAB_];   // double-buffered TDM landing pad (64.5 KB)
  __shared__ float A[I_*D_*W_];      // x_avg[i][d][w]  (K-averaged capsules)
  __shared__ float Nrm[I_*W_];       // ||x_avg|| over d
  __shared__ float V[D_*W_];         // consensus vector per pixel
  __shared__ float L[I_*W_];         // routing losses
  __shared__ float THR[W_];          // 26th-smallest loss per pixel

  const int tid = threadIdx.x;
  const int blk = blockIdx.x;        // one block per (b, o, h)
  const int b   = blk / (O_*H_);
  const int rem = blk % (O_*H_);
  const int o   = rem / H_;
  const int h   = rem % H_;

  // element offset of x[b, i=0, o, d=0, h, w=0, 0, 0]
  const size_t base0 =
      ((size_t)((b*I_)*O_ + o) * (size_t)(D_*H_) + (size_t)h) * (size_t)WK_;

  // ---- TDM pipeline: prefetch chunk 0 ----
  if (tid == 0) {
    tdm_load_tile(x + base0, (u32)(uintptr_t)&slab[0][0]);
  }

  for (int ch = 0; ch < NCH_; ++ch) {
    if (tid == 0) {
      if (ch + 1 < NCH_) {
        tdm_load_tile(x + base0 + (size_t)(ch + 1) * CHI_ * (size_t)STRIDE_I_,
                      (u32)(uintptr_t)&slab[(ch + 1) & 1][0]);
        __builtin_amdgcn_s_wait_tensorcnt(1);   // chunk ch landed; ch+1 in flight
      } else {
        __builtin_amdgcn_s_wait_tensorcnt(0);
      }
    }
    __syncthreads();   // slab[ch&1] is ready for everyone

    // Phase A: per (i, w): norm-weighted average over the K=9 kernel taps.
    // Both passes run k-innermost so each LDS access is a contiguous 9-float
    // run (vectorizes to ds_load_b128/b96; lane stride 9 words, gcd(9,64)=1
    // -> bank-conflict-free).
    if (tid < CHI_*W_) {
      const int w  = tid % W_;
      const int il = tid / W_;
      const int i  = ch*CHI_ + il;
      const float* s = &slab[ch & 1][0];

      float xn2[K_];
      #pragma unroll
      for (int k = 0; k < K_; ++k) xn2[k] = 0.f;
      #pragma unroll
      for (int d = 0; d < D_; ++d) {
        const float* row = &s[(il*D_ + d)*WK_ + w*K_];
        #pragma unroll
        for (int k = 0; k < K_; ++k) {
          float v = row[k];
          xn2[k] += v*v;
        }
      }
      float xn[K_];
      float wsum = 0.f;
      #pragma unroll
      for (int k = 0; k < K_; ++k) {
        xn[k] = sqrtf(xn2[k]);
        wsum += xn[k];
      }
      const float inv = 1.f / wsum;
      float n2 = 0.f;
      #pragma unroll
      for (int d = 0; d < D_; ++d) {
        const float* row = &s[(il*D_ + d)*WK_ + w*K_];
        float acc = 0.f;
        #pragma unroll
        for (int k = 0; k < K_; ++k)
          acc += xn[k] * row[k];
        acc *= inv;
        A[(i*D_ + d)*W_ + w] = acc;
        n2 += acc*acc;
      }
      Nrm[i*W_ + w] = sqrtf(n2);
    }
    __syncthreads();   // chunk ch consumed; its buffer may be overwritten next iter
  }

  // Phase B: consensus v[d][w] = sum_i n_i * A / sum_i n_i
  if (tid < D_*W_) {
    const int w = tid % W_;
    const int d = tid / W_;
    float num = 0.f, den = 0.f;
    #pragma unroll
    for (int i = 0; i < I_; ++i) {
      float n = Nrm[i*W_ + w];
      num += n * A[(i*D_ + d)*W_ + w];
      den += n;
    }
    V[d*W_ + w] = num / den;
  }
  __syncthreads();

  // Phase C: losses[i][w] = -<v, x_avg_i>
  for (int t = tid; t < I_*W_; t += 256) {
    const int w = t % W_;
    const int i = t / W_;
    float acc = 0.f;
    #pragma unroll
    for (int d = 0; d < D_; ++d)
      acc += V[d*W_ + w] * A[(i*D_ + d)*W_ + w];
    L[i*W_ + w] = -acc;
  }
  __syncthreads();

  // Phase D: 26th-smallest loss over the 32 input caps (counting selection;
  // tie handling matches sort()[25])
  if (tid < W_) {
    const int w = tid;
    float thr = 0.f;
    for (int j = 0; j < I_; ++j) {
      float vj = L[j*W_ + w];
      int c = 0, e = 0;
      for (int l = 0; l < I_; ++l) {
        float vl = L[l*W_ + w];
        c += (vl <  vj);
        e += (vl == vj);
      }
      if (c <= SUBSET_-1 && c + e >= SUBSET_) thr = vj;
    }
    THR[w] = thr;
  }
  __syncthreads();

  // Phase E: masked norm-weighted consensus -> output [B,O,D,H,W]
  if (tid < D_*W_) {
    const int w = tid % W_;
    const int d = tid / W_;
    const float thr = THR[w];
    float num = 0.f, den = 0.f;
    #pragma unroll
    for (int i = 0; i < I_; ++i) {
      float m = (L[i*W_ + w] <= thr) ? Nrm[i*W_ + w] : 0.f;
      num += m * A[(i*D_ + d)*W_ + w];
      den += m;
    }
    out[((((size_t)b*O_ + o)*D_ + d)*H_ + h)*W_ + w] = num / den;
  }
}

extern "C" void kernel_launch(void* const* d_in, const int* in_sizes, int n_in,
                              void* d_out, int out_size, void* d_ws, size_t ws_size,
                              hipStream_t stream) {
  const float* x = (const float*)d_in[0];
  float* out = (float*)d_out;
  (void)in_sizes; (void)n_in; (void)out_size; (void)d_ws; (void)ws_size;
  ConvSubsetRouting_76699525972648_kernel<<<dim3(B_*O_*H_), dim3(256), 0, stream>>>(x, out);
}